// vector_unpack_46608985096504
// MI455X (gfx1250) — compile-verified
//
#include <hip/hip_runtime.h>
#include <hip/hip_bf16.h>

typedef __attribute__((ext_vector_type(2))) float v2f;
typedef __attribute__((ext_vector_type(4))) float v4f;
typedef __attribute__((ext_vector_type(8))) float v8f;

#define B_ 64
#define T_ 512
#define D_ 1024
#define THALF_ (T_ / 2)

// ---------------------------------------------------------------------------
// Kernel 0: w[b,t] = (t < len[b]) ? weights[word[b,t]] : 0     (tiny gather)
// ---------------------------------------------------------------------------
__global__ __launch_bounds__(256) void wtab_kernel(
    const int* __restrict__ wordseq, const int* __restrict__ slen,
    const float* __restrict__ weights, float* __restrict__ wtab) {
  int i = blockIdx.x * 256 + threadIdx.x;       // over B*T = 32768
  int b = i >> 9;                                // /T_
  int t = i & (T_ - 1);
  float w = 0.0f;
  if (t < slen[b]) w = weights[wordseq[i]];
  wtab[i] = w;
}

// ---------------------------------------------------------------------------
// Kernel 1: streaming WMMA reduction over T.
// One wave owns (b, 64 d-columns, half of T). Per K-step of 4 t's:
//   A (16x4 f32): row0 = mask(t), row1 = w(t), rows 2..15 = 0
//   B (4x16 f32) x4 tiles from two b128 loads per lane (d interleaved by 4)
//   C (16x16 f32): row0 accumulates s, row1 accumulates y_hat
// A-layout (ISA 7.12.2, 32-bit A 16x4): lane m=l&15, half h=l>>4;
//   vgpr0 = A[m][2h], vgpr1 = A[m][2h+1].  B mirrors with N=lane&15.
// T-half offset is folded into the pointers so the loop has a
// compile-time-constant trip count: scalar loop, EXEC all-1s around WMMA.
// ---------------------------------------------------------------------------
__global__ __launch_bounds__(256) void fused_wmma_kernel(
    const float* __restrict__ vs, const int* __restrict__ slen,
    const float* __restrict__ wtab,
    float* __restrict__ s_part, float* __restrict__ yh_part) {
  __shared__ float lds_w[T_];

  const int tid  = threadIdx.x;
  const int lane = tid & 31;
  const int wid  = blockIdx.x * 8 + (tid >> 5);  // 2048 waves total
  const int b      = wid >> 5;                   // 32 waves per batch row
  const int tile   = (wid >> 1) & 15;            // 16 d-tiles of 64 columns
  const int thalf  = wid & 1;                    // T split in two halves
  const int d_base = tile * 64;
  const int tbase  = thalf * THALF_;

  // whole block shares one b -> stage w row once in LDS (2 KB)
  for (int i = tid; i < T_; i += 256) lds_w[i] = wtab[b * T_ + i];
  __syncthreads();

  const int   m    = lane & 15;
  const int   h    = lane >> 4;
  const int   lenl = slen[b] - tbase;            // len relative to this half
  const float rowsel0 = (m == 0) ? 1.0f : 0.0f;  // picks mask row
  const float rowsel1 = (m == 1) ? 1.0f : 0.0f;  // picks w row

  v8f c0 = {}, c1 = {}, c2 = {}, c3 = {};

  // lane's 4 d-columns: d_base + 4*m .. +3 ; rows advance by D_
  const float* base  = vs + (size_t)b * T_ * D_ + (size_t)tbase * D_ +
                       (size_t)(d_base + 4 * m);
  const float* wrow  = lds_w + tbase;

#pragma unroll 8
  for (int t0 = 0; t0 < THALF_; t0 += 4) {
    const int te = t0 + 2 * h;                       // this half-wave's even row
    v4f r0 = *(const v4f*)(base + (size_t)te * D_);        // K = 2h
    v4f r1 = *(const v4f*)(base + (size_t)(te + 1) * D_);  // K = 2h+1

    const float mk0 = (te     < lenl) ? 1.0f : 0.0f;
    const float mk1 = (te + 1 < lenl) ? 1.0f : 0.0f;
    const float w0  = wrow[te];
    const float w1  = wrow[te + 1];
    v2f a;
    a.x = rowsel0 * mk0 + rowsel1 * w0;   // rows >=2 get 0
    a.y = rowsel0 * mk1 + rowsel1 * w1;

    v2f bop;
    bop.x = r0.x; bop.y = r1.x;
    c0 = __builtin_amdgcn_wmma_f32_16x16x4_f32(false, a, false, bop, (short)0, c0, false, false);
    bop.x = r0.y; bop.y = r1.y;
    c1 = __builtin_amdgcn_wmma_f32_16x16x4_f32(false, a, false, bop, (short)0, c1, false, false);
    bop.x = r0.z; bop.y = r1.z;
    c2 = __builtin_amdgcn_wmma_f32_16x16x4_f32(false, a, false, bop, (short)0, c2, false, false);
    bop.x = r0.w; bop.y = r1.w;
    c3 = __builtin_amdgcn_wmma_f32_16x16x4_f32(false, a, false, bop, (short)0, c3, false, false);
  }

  // C rows 0 (s) and 1 (y_hat) live in vgpr0/vgpr1 of lanes 0..15, N = lane.
  // Tile j's column n is d_base + 4n + j  ->  contiguous float4 per lane.
  if (lane < 16) {
    v4f sv, hv;
    sv.x = c0[0]; sv.y = c1[0]; sv.z = c2[0]; sv.w = c3[0];
    hv.x = c0[1]; hv.y = c1[1]; hv.z = c2[1]; hv.w = c3[1];
    const size_t off = (size_t)(thalf * B_ + b) * D_ + d_base + 4 * lane;
    *(v4f*)(s_part  + off) = sv;
    *(v4f*)(yh_part + off) = hv;
  }
}

// ---------------------------------------------------------------------------
// Kernel 2: combine the two T-half partials, block-reduce L1 norm per b,
//           emit y = s / sqrt(||s||_1) and y_hat. One block per batch row.
// ---------------------------------------------------------------------------
__global__ __launch_bounds__(256) void finalize_kernel(
    const float* __restrict__ s_part, const float* __restrict__ yh_part,
    float* __restrict__ y_out, float* __restrict__ yh_out) {
  __shared__ float red[256];
  const int b   = blockIdx.x;
  const int tid = threadIdx.x;

  const v4f* s0 = (const v4f*)(s_part + (size_t)b * D_);
  const v4f* s1 = (const v4f*)(s_part + (size_t)(B_ + b) * D_);
  const v4f* h0 = (const v4f*)(yh_part + (size_t)b * D_);
  const v4f* h1 = (const v4f*)(yh_part + (size_t)(B_ + b) * D_);

  v4f sv = s0[tid];
  v4f sw = s1[tid];
  sv.x += sw.x; sv.y += sw.y; sv.z += sw.z; sv.w += sw.w;

  v4f hv = h0[tid];
  v4f hw = h1[tid];
  hv.x += hw.x; hv.y += hw.y; hv.z += hw.z; hv.w += hw.w;
  ((v4f*)(yh_out + (size_t)b * D_))[tid] = hv;

  red[tid] = fabsf(sv.x) + fabsf(sv.y) + fabsf(sv.z) + fabsf(sv.w);
  __syncthreads();
  for (int off = 128; off > 0; off >>= 1) {
    if (tid < off) red[tid] += red[tid + off];
    __syncthreads();
  }
  const float inv = 1.0f / sqrtf(red[0]);
  v4f yv;
  yv.x = sv.x * inv; yv.y = sv.y * inv; yv.z = sv.z * inv; yv.w = sv.w * inv;
  ((v4f*)(y_out + (size_t)b * D_))[tid] = yv;
}

// ---------------------------------------------------------------------------
extern "C" void kernel_launch(void* const* d_in, const int* in_sizes, int n_in,
                              void* d_out, int out_size, void* d_ws, size_t ws_size,
                              hipStream_t stream) {
  const float* vs      = (const float*)d_in[0];  // [B,T,D] f32
  const int*   slen    = (const int*)d_in[1];    // [B] i32
  const int*   wordseq = (const int*)d_in[2];    // [B,T] i32
  const float* weights = (const float*)d_in[3];  // [VOCAB] f32

  float* y_out  = (float*)d_out;            // [B,D]
  float* yh_out = (float*)d_out + B_ * D_;  // [B,D]

  float* wtab    = (float*)d_ws;            // [B,T]       128 KB
  float* s_part  = wtab + B_ * T_;          // [2,B,D]     512 KB
  float* yh_part = s_part + 2 * B_ * D_;    // [2,B,D]     512 KB

  wtab_kernel<<<(B_ * T_) / 256, 256, 0, stream>>>(wordseq, slen, weights, wtab);
  fused_wmma_kernel<<<256, 256, 0, stream>>>(vs, slen, wtab, s_part, yh_part);
  finalize_kernel<<<B_, 256, 0, stream>>>(s_part, yh_part, y_out, yh_out);
}